// TopoTransform_10763188043933
// MI455X (gfx1250) — compile-verified
//
#include <hip/hip_runtime.h>

// CDNA5 / gfx1250, wave32. Blocked left-looking QR (panel width 16) with the
// O(C^3) projection work on V_WMMA_F32_16X16X4_F32, plus a WMMA apply GEMM.
// Whole 256x256 f32 matrix + panel scratch lives in LDS (303KB < 320KB/WG).

typedef __attribute__((ext_vector_type(2))) float v2f;
typedef __attribute__((ext_vector_type(8))) float v8f;

#define NT 256  // 8 wave32s

// One workgroup per spatial site s.
//   W: (S, C, C) row-major  -> Q[s] (diag(R) > 0, matches sign-fixed QR)
//   X: (32, C, S), Y: (32, C, S),  Y[b,:,s] = Q[s] @ X[b,:,s]
template <int C>
__global__ __launch_bounds__(NT) void blocked_qr_mix_kernel(
    const float* __restrict__ W, const float* __restrict__ X,
    float* __restrict__ Y, int S) {
  const int s    = blockIdx.x;
  const int tid  = threadIdx.x;
  const int lane = tid & 31;
  const int half = lane >> 4;      // WMMA half-wave (K pair / row half)
  const int l    = lane & 15;
  const int wave = tid >> 5;
  constexpr int NW = NT / 32;      // 8 waves
  constexpr int MT = C / 16;       // 16-wide tiles along C

  extern __shared__ float lds[];
  float* Ws = lds;                 // C*C row-major; columns are the QR vectors
  float* Xs = Ws + C * C;          // C*32 as (c, b)
  float* Ts = Xs + C * 32;         // (C-16)*16: T = Qprev^T @ Vpanel
  float* rc = Ts + (C - 16) * 16;  // 16: in-panel dot coefficients

  // ---- stage W[s] and X[:, :, s] into LDS ----
  const float* Wm = W + (size_t)s * C * C;
  for (int i = tid; i < C * C; i += NT) Ws[i] = Wm[i];
  for (int i = tid; i < C * 32; i += NT) {
    int c = i >> 5, b = i & 31;
    Xs[i] = X[(size_t)b * C * S + (size_t)c * S + s];
  }
  __syncthreads();

  // ---- blocked left-looking QR, panels of 16 columns ----
  for (int p = 0; p < MT; ++p) {
    const int c0 = p * 16;
    const int J  = c0;  // # previously orthogonalized columns

    // Project panel against Q_prev: 2 CGS passes, all WMMA.
    for (int pass = 0; pass < 2 && p > 0; ++pass) {
      // GEMM1: T (J x 16) = Qprev^T (J x C) @ Vp (C x 16)
      //   A[m][k] = Ws[k*C + m]  (transposed read),  B[k][n] = Ws[k*C + c0 + n]
      for (int mt = wave; mt < p; mt += NW) {
        v8f acc = {};
        for (int k0 = 0; k0 < C; k0 += 4) {
          const int ka = k0 + 2 * half;
          v2f a, b;
          a.x = Ws[ka * C + mt * 16 + l];
          a.y = Ws[(ka + 1) * C + mt * 16 + l];
          b.x = Ws[ka * C + c0 + l];
          b.y = Ws[(ka + 1) * C + c0 + l];
          acc = __builtin_amdgcn_wmma_f32_16x16x4_f32(
              false, a, false, b, (short)0, acc, false, false);
        }
#pragma unroll
        for (int r = 0; r < 8; ++r)
          Ts[(mt * 16 + r + 8 * half) * 16 + l] = acc[r];
      }
      __syncthreads();
      // GEMM2: Vp -= Qprev (C x J) @ T (J x 16)
      //   A[m][k] = Ws[m*C + k] (contiguous -> b64 LDS loads), B[k][n] = Ts[k*16+n]
      for (int mt = wave; mt < MT; mt += NW) {
        v8f acc = {};
        for (int k0 = 0; k0 < J; k0 += 4) {
          const int ka = k0 + 2 * half;
          v2f a = *(const v2f*)&Ws[(mt * 16 + l) * C + ka];
          v2f b;
          b.x = Ts[ka * 16 + l];
          b.y = Ts[(ka + 1) * 16 + l];
          acc = __builtin_amdgcn_wmma_f32_16x16x4_f32(
              false, a, false, b, (short)0, acc, false, false);
        }
#pragma unroll
        for (int r = 0; r < 8; ++r)
          Ws[(mt * 16 + r + 8 * half) * C + c0 + l] -= acc[r];  // f32 WMMA has no A-neg
      }
      __syncthreads();
    }

    // In-panel MGS over 16 columns.
    //   16 half-wave groups compute all trailing dots concurrently (shfl
    //   reduce, no barrier); then each thread owns whole rows for the fused
    //   normalize + rank-1 update (race-free).
    const int grp = tid >> 4;  // panel-local column this group handles
    const int gr  = tid & 15;
    for (int jl = 0; jl < 16; ++jl) {
      const int j = c0 + jl;
      __syncthreads();  // previous update visible
      float d = 0.0f;
      if (grp >= jl) {
        for (int i = gr; i < C; i += 16)
          d += Ws[i * C + j] * Ws[i * C + c0 + grp];
      }
#pragma unroll
      for (int m = 8; m >= 1; m >>= 1) d += __shfl_xor(d, m, 16);
      if (gr == 0) rc[grp] = d;
      __syncthreads();
      const float rinv  = 1.0f / sqrtf(rc[jl] + 1e-30f);  // r_jj > 0
      const float scale = rinv * rinv;  // coef vs UNnormalized v_j
      for (int i = tid; i < C; i += NT) {
        const float v = Ws[i * C + j];
        for (int k = jl + 1; k < 16; ++k)
          Ws[i * C + c0 + k] -= (rc[k] * scale) * v;
        Ws[i * C + j] = v * rinv;
      }
    }
    __syncthreads();
  }

  // ---- apply: Y_s (C x 32) = Q (C x C) @ X_s (C x 32) via WMMA ----
  for (int t = wave; t < MT * 2; t += NW) {  // uniform per wave -> EXEC all-1s
    const int m = t >> 1;
    const int n = t & 1;
    v8f acc = {};
    for (int k0 = 0; k0 < C; k0 += 4) {
      const int ka = k0 + 2 * half;
      v2f a = *(const v2f*)&Ws[(m * 16 + l) * C + ka];
      v2f b;
      b.x = Xs[ka * 32 + n * 16 + l];
      b.y = Xs[(ka + 1) * 32 + n * 16 + l];
      acc = __builtin_amdgcn_wmma_f32_16x16x4_f32(
          false, a, false, b, (short)0, acc, false, false);
    }
#pragma unroll
    for (int r = 0; r < 8; ++r) {
      const int c  = m * 16 + r + 8 * half;
      const int bb = n * 16 + l;
      Y[(size_t)bb * C * S + (size_t)c * S + s] = acc[r];
    }
  }
}

extern "C" void kernel_launch(void* const* d_in, const int* in_sizes, int n_in,
                              void* d_out, int out_size, void* d_ws,
                              size_t ws_size, hipStream_t stream) {
  (void)in_sizes; (void)n_in; (void)d_ws; (void)ws_size; (void)out_size;
  const float* x0 = (const float*)d_in[0];  // (32, 64, 28, 28)
  const float* x1 = (const float*)d_in[1];  // (32, 256, 14, 14)
  const float* W0 = (const float*)d_in[2];  // (784, 64, 64)
  const float* W1 = (const float*)d_in[3];  // (196, 256, 256)

  float* y0 = (float*)d_out;
  float* y1 = y0 + (size_t)32 * 64 * 28 * 28;

  const int S0 = 28 * 28;  // 784
  const int S1 = 14 * 14;  // 196

  const size_t lds0 =
      (size_t)(64 * 64 + 64 * 32 + (64 - 16) * 16 + 16) * sizeof(float);
  const size_t lds1 =
      (size_t)(256 * 256 + 256 * 32 + (256 - 16) * 16 + 16) * sizeof(float);
  // lds1 ~= 303 KB: legal on CDNA5 (up to 320 KB LDS per workgroup).

  blocked_qr_mix_kernel<64><<<S0, NT, lds0, stream>>>(W0, x0, y0, S0);
  blocked_qr_mix_kernel<256><<<S1, NT, lds1, stream>>>(W1, x1, y1, S1);
}